// iterativeGCN_variant_4269197492791
// MI455X (gfx1250) — compile-verified
//
#include <hip/hip_runtime.h>
#include <cstdint>

// ---------------------------------------------------------------------------
// Iterative GCN for MI455X (gfx1250, wave32).
//  - Dense node transforms use V_WMMA_F32_16X16X4_F32 (exact fp32 numerics).
//  - W is staged per-block into LDS pre-permuted to B-fragment layout:
//    inner loop = 1 global_load_b64 (A) + 8 ds_load_b64 (B) + 8 WMMA.
//  - Edge aggregation uses hardware global_atomic_add_f32 (L2-resident: the
//    51MB feature matrix fits the 192MB L2, so scatter/gather hit L2).
// ---------------------------------------------------------------------------

typedef float v2f __attribute__((ext_vector_type(2)));
typedef float v8f __attribute__((ext_vector_type(8)));

#if defined(__HIP_DEVICE_COMPILE__)
#if !__has_builtin(__builtin_amdgcn_wmma_f32_16x16x4_f32)
#error "missing __builtin_amdgcn_wmma_f32_16x16x4_f32 on this toolchain"
#endif
#endif

// ---------------- degree / normalization ----------------
__global__ void k_init_deg(float* deg, int n) {
  int i = blockIdx.x * blockDim.x + threadIdx.x;
  if (i < n) deg[i] = 1.0f;  // self-loop contributes 1 to every node's degree
}

__global__ void k_count_deg(const int* __restrict__ dst, float* deg, int E) {
  int e = blockIdx.x * blockDim.x + threadIdx.x;
  if (e < E) unsafeAtomicAdd(&deg[dst[e]], 1.0f);
}

__global__ void k_make_dis(const float* __restrict__ deg, float* dis, int n) {
  int i = blockIdx.x * blockDim.x + threadIdx.x;
  if (i < n) dis[i] = rsqrtf(deg[i]);  // deg >= 1 always (self loops)
}

// ---------------- WMMA fp32 GEMM: H = A @ W, OUT0 = H * dis^2 --------------
// One wave computes a 16-row strip across all DOUT columns.
// A fragment (16x4, f32): lane L -> row M=L%16, K = 2*(L/16)+v  (v2f load)
// B fragment (4x16, f32): lane L -> col N=L%16, K = 2*(L/16)+v  (from LDS)
// C/D (16x16, f32, 8 VGPR): vgpr r -> M = r + 8*(L/16), N = L%16
// W is staged into LDS in fragment order: frag (kb, t), lane L holds the
// two floats W[(4kb + 2*(L/16) + v)*DOUT + 16t + (L%16)], v = 0,1 -> one
// bank-conflict-free ds_load_b64 per fragment.
template <int KDIM, int DOUT>
__global__ __launch_bounds__(256) void gemm_wmma_kernel(
    const float* __restrict__ A, const float* __restrict__ W,
    const float* __restrict__ dis, float* __restrict__ H,
    float* __restrict__ OUT0, int nRows) {
  constexpr int NT = DOUT / 16;            // 16-col tiles
  constexpr int CHUNK_K = 128;             // K staged per LDS pass (<=64KB)
  constexpr int KCHUNKS = KDIM / CHUNK_K;  // 2 for enc, else 1
  constexpr int KBC = CHUNK_K / 4;         // k-steps per chunk
  constexpr int NFRAG = KBC * NT;          // fragments per chunk
  extern __shared__ float ldsW[];          // NFRAG * 64 floats

  const int lane = threadIdx.x & 31;
  const int wave = threadIdx.x >> 5;
  const int g    = lane >> 4;   // half-wave select
  const int ln   = lane & 15;   // 0..15
  const int rowBase = (blockIdx.x * 8 + wave) * 16;
  const bool active = (rowBase < nRows);   // wave-uniform

  v8f acc[NT];
#pragma unroll
  for (int t = 0; t < NT; ++t) acc[t] = (v8f){};

  const int arow = min(rowBase + ln, nRows - 1);  // clamp tail rows
  const float* Arow = A + (size_t)arow * KDIM;

  for (int kc = 0; kc < KCHUNKS; ++kc) {
    const int ko = kc * CHUNK_K;
    // ---- stage W[ko:ko+CHUNK_K, :] into LDS in fragment order ----
    for (int idx = threadIdx.x; idx < NFRAG * 32; idx += 256) {
      int l    = idx & 31;
      int frag = idx >> 5;
      int kb   = frag / NT;
      int t    = frag - kb * NT;
      int k    = ko + 4 * kb + 2 * (l >> 4);
      int n    = t * 16 + (l & 15);
      ldsW[idx * 2 + 0] = W[(size_t)k * DOUT + n];
      ldsW[idx * 2 + 1] = W[(size_t)(k + 1) * DOUT + n];
    }
    __syncthreads();

    if (active) {  // wave-uniform: EXEC all-ones around WMMA
#pragma unroll 4
      for (int kb = 0; kb < KBC; ++kb) {
        v2f a = *(const v2f*)(Arow + ko + 4 * kb + 2 * g);
#pragma unroll
        for (int t = 0; t < NT; ++t) {
          v2f b = *(const v2f*)(&ldsW[((kb * NT + t) * 32 + lane) * 2]);
          acc[t] = __builtin_amdgcn_wmma_f32_16x16x4_f32(
              /*neg_a=*/false, a, /*neg_b=*/false, b,
              /*c_mod=*/(short)0, acc[t], /*reuse_a=*/false, /*reuse_b=*/false);
        }
      }
    }
    if (KCHUNKS > 1) __syncthreads();  // protect LDS before next refill
  }

  if (active) {
#pragma unroll
    for (int r = 0; r < 8; ++r) {
      int row = rowBase + r + 8 * g;
      if (row < nRows) {
        float dv = dis[row];
        float self = dv * dv;  // self-loop norm = dis[n]*dis[n]
#pragma unroll
        for (int t = 0; t < NT; ++t) {
          float v = acc[t][r];
          size_t idx = (size_t)row * DOUT + t * 16 + ln;
          H[idx] = v;
          OUT0[idx] = v * self;
        }
      }
    }
  }
}

// ---------------- edge scatter: OUT[dst] += H[src]*dis[src]*dis[dst] -------
// D=128: one wave per edge, lane = float4 chunk -> fully coalesced gather.
template <int D>
__global__ __launch_bounds__(256) void k_edge_scatter(
    const float* __restrict__ H, const int* __restrict__ src,
    const int* __restrict__ dst, const float* __restrict__ dis,
    float* __restrict__ OUT, int E) {
  constexpr int C = D / 4;  // float4 chunks per edge
  long tid = (long)blockIdx.x * blockDim.x + threadIdx.x;
  long e = tid / C;
  int  c = (int)(tid % C);
  if (e >= E) return;
  int s = src[e];
  int t = dst[e];
  float norm = dis[s] * dis[t];
  const float4 hv = *(const float4*)(H + (size_t)s * D + 4 * c);
  float* o = OUT + (size_t)t * D + 4 * c;
  unsafeAtomicAdd(o + 0, hv.x * norm);
  unsafeAtomicAdd(o + 1, hv.y * norm);
  unsafeAtomicAdd(o + 2, hv.z * norm);
  unsafeAtomicAdd(o + 3, hv.w * norm);
}

// ---------------- epilogues ----------------
__global__ void k_finish_relu(const float* __restrict__ OUT,
                              const float* __restrict__ b, float* __restrict__ h,
                              int total, int colMask) {
  int i = blockIdx.x * blockDim.x + threadIdx.x;
  if (i >= total) return;
  float v = OUT[i] + b[i & colMask];
  h[i] = fmaxf(v, 0.0f);
}

__global__ void k_finish_blend(const float* __restrict__ OUT,
                               const float* __restrict__ b, float* __restrict__ h,
                               float sf, int total, int colMask) {
  int i = blockIdx.x * blockDim.x + threadIdx.x;
  if (i >= total) return;
  float v = fmaxf(OUT[i] + b[i & colMask], 0.0f);
  h[i] = sf * h[i] + (1.0f - sf) * v;
}

__global__ void k_finish_bias(const float* __restrict__ OUT,
                              const float* __restrict__ b, float* __restrict__ out,
                              int total, int colMask) {
  int i = blockIdx.x * blockDim.x + threadIdx.x;
  if (i >= total) return;
  out[i] = OUT[i] + b[i & colMask];
}

// ---------------- launcher ----------------
extern "C" void kernel_launch(void* const* d_in, const int* in_sizes, int n_in,
                              void* d_out, int out_size, void* d_ws, size_t ws_size,
                              hipStream_t stream) {
  const float* x     = (const float*)d_in[0];
  const int*   eidx  = (const int*)d_in[1];
  const float* W_enc = (const float*)d_in[2];
  const float* b_enc = (const float*)d_in[3];
  const float* W_gc  = (const float*)d_in[4];
  const float* b_gc  = (const float*)d_in[5];
  const float* W_dec = (const float*)d_in[6];
  const float* b_dec = (const float*)d_in[7];

  const int N = in_sizes[0] / 256;  // D_IN = 256
  const int E = in_sizes[1] / 2;    // edge_index is [2, E]
  const int* src = eidx;
  const int* dst = eidx + E;

  auto a256 = [](size_t v) { return (v + 255) & ~(size_t)255; };
  char* p = (char*)d_ws;
  float* deg = (float*)p; p += a256((size_t)N * 4);
  float* dis = (float*)p; p += a256((size_t)N * 4);
  float* h   = (float*)p; p += a256((size_t)N * 128 * 4);
  float* Hb  = (float*)p; p += a256((size_t)N * 128 * 4);
  float* Ob  = (float*)p; p += a256((size_t)N * 128 * 4);
  (void)ws_size; (void)n_in; (void)out_size;

  const int TB = 256;
  auto cdiv = [](long a, long b) { return (int)((a + b - 1) / b); };

  const size_t shW128 = 128 * 128 * sizeof(float);  // 64KB staging
  const size_t shW64  = 128 * 64 * sizeof(float);   // 32KB staging

  // normalization
  k_init_deg <<<cdiv(N, TB), TB, 0, stream>>>(deg, N);
  k_count_deg<<<cdiv(E, TB), TB, 0, stream>>>(dst, deg, E);
  k_make_dis <<<cdiv(N, TB), TB, 0, stream>>>(deg, dis, N);

  // encoder: h = relu(conv(x, W_enc, b_enc))
  gemm_wmma_kernel<256, 128><<<cdiv(N, 128), 256, shW128, stream>>>(x, W_enc, dis, Hb, Ob, N);
  k_edge_scatter<128><<<cdiv((long)E * 32, TB), TB, 0, stream>>>(Hb, src, dst, dis, Ob, E);
  k_finish_relu<<<cdiv((long)N * 128, TB), TB, 0, stream>>>(Ob, b_enc, h, N * 128, 127);

  // 8 smoothing iterations: h = 0.7*h + 0.3*relu(conv(h, W_gc, b_gc))
  for (int it = 0; it < 8; ++it) {
    gemm_wmma_kernel<128, 128><<<cdiv(N, 128), 256, shW128, stream>>>(h, W_gc, dis, Hb, Ob, N);
    k_edge_scatter<128><<<cdiv((long)E * 32, TB), TB, 0, stream>>>(Hb, src, dst, dis, Ob, E);
    k_finish_blend<<<cdiv((long)N * 128, TB), TB, 0, stream>>>(Ob, b_gc, h, 0.7f, N * 128, 127);
  }

  // decoder: out = conv(h, W_dec, b_dec)
  gemm_wmma_kernel<128, 64><<<cdiv(N, 128), 256, shW64, stream>>>(h, W_dec, dis, Hb, Ob, N);
  k_edge_scatter<64><<<cdiv((long)E * 16, TB), TB, 0, stream>>>(Hb, src, dst, dis, Ob, E);
  k_finish_bias<<<cdiv((long)N * 64, TB), TB, 0, stream>>>(Ob, b_dec, (float*)d_out, N * 64, 63);
}